// GCN_68418829025864
// MI455X (gfx1250) — compile-verified
//
#include <hip/hip_runtime.h>

typedef __attribute__((ext_vector_type(2))) float v2f;
typedef __attribute__((ext_vector_type(8))) float v8f;

#define NNODES 50000
#define NEDGES 1600000
#define KT 64     // K-chunk per LDS stage
#define PSTR 80   // pair-row stride in float2: 160 words mod 64 banks = 32 -> half-waves disjoint

__device__ __forceinline__ void atomAddF(float* p, float v) {
    unsafeAtomicAdd(p, v);   // lowers to global_atomic_add_f32 (no CAS loop)
}

// ---------- normalization precompute ----------
__global__ void k_init_deg(float* deg, int n) {
    int i = blockIdx.x * blockDim.x + threadIdx.x;
    if (i < n) deg[i] = 1.0f;            // self-loop contribution
}

__global__ void k_count_deg(const long long* __restrict__ ei, int* __restrict__ row32,
                            int* __restrict__ col32, float* deg, int e_cnt) {
    int e = blockIdx.x * blockDim.x + threadIdx.x;
    if (e >= e_cnt) return;
    int r = (int)ei[e];
    int c = (int)ei[e_cnt + e];
    row32[e] = r;
    col32[e] = c;
    atomAddF(deg + c, 1.0f);
}

__global__ void k_dinv(float* deg, int n) {
    int i = blockIdx.x * blockDim.x + threadIdx.x;
    if (i < n) deg[i] = rsqrtf(deg[i]);  // deg >= 1 always (self loops)
}

__global__ void k_norm(const int* __restrict__ row, const int* __restrict__ col,
                       const float* __restrict__ dinv, float* __restrict__ nrm, int e_cnt) {
    int e = blockIdx.x * blockDim.x + threadIdx.x;
    if (e >= e_cnt) return;
    nrm[e] = dinv[row[e]] * dinv[col[e]];
}

// ---------- dense transform: T[M,NO] = X[M,K] @ W[K,NO], fp32 WMMA ----------
// Block: 8 waves -> 128 rows x 64 cols. B panel staged through LDS in 64-k chunks,
// stored as K-PAIRED float2 so each B fragment is one conflict-free ds_load_b64
// landing directly in the even-aligned VGPR pair WMMA consumes (no shuffles).
__global__ void __launch_bounds__(256) k_gemm_wmma(
        const float* __restrict__ X, const float* __restrict__ W,
        float* __restrict__ T, int M, int K, int NO) {
    __shared__ float2 bs2[(KT / 2) * PSTR];   // 32 pair-rows x 80 cols = 20 KB

    const int t    = threadIdx.x;
    const int lane = t & 31;
    const int wave = t >> 5;
    const int MT   = M >> 4;
    const int mt   = blockIdx.x * 8 + wave;
    const int mtile = (mt < MT) ? mt : (MT - 1);   // clamp: all waves join barriers
    const int n0   = blockIdx.y * 64;
    const int half = lane >> 4;          // 0: K pair {0,1}, 1: K pair {2,3}
    const int mr   = lane & 15;          // M row (A) / N col (B, C)

    const float* A = X + (size_t)(mtile * 16 + mr) * K + 2 * half;

    // cooperative B staging: thread owns column sn, pair-rows 8*spb .. 8*spb+7
    const int sn  = t & 63;              // column within 64-wide panel
    const int spb = t >> 6;              // 0..3
    const float* Wc = W + n0 + sn;

    float sa[8], sb[8];
#pragma unroll
    for (int j = 0; j < 8; ++j) {
        int p = spb * 8 + j;
        sa[j] = Wc[(size_t)(2 * p) * NO];
        sb[j] = Wc[(size_t)(2 * p + 1) * NO];
    }

    v8f acc0 = {}, acc1 = {}, acc2 = {}, acc3 = {};
    for (int k0 = 0; k0 < K; k0 += KT) {
#pragma unroll
        for (int j = 0; j < 8; ++j)
            bs2[(spb * 8 + j) * PSTR + sn] = make_float2(sa[j], sb[j]);
        __syncthreads();

        if (k0 + KT < K) {               // register double-buffer: overlap with compute
#pragma unroll
            for (int j = 0; j < 8; ++j) {
                int p = spb * 8 + j;
                sa[j] = Wc[(size_t)(k0 + KT + 2 * p) * NO];
                sb[j] = Wc[(size_t)(k0 + KT + 2 * p + 1) * NO];
            }
            __builtin_prefetch(A + k0 + KT, 0, 1);   // global_prefetch_b8 for next A chunk
        }

        const float* Ak = A + k0;
#pragma unroll 4
        for (int kk = 0; kk < KT; kk += 4) {
            v2f a = *(const v2f*)(Ak + kk);
            const float2* L = &bs2[((kk >> 1) + half) * PSTR + mr];
            v2f b0 = *(const v2f*)(L + 0);
            v2f b1 = *(const v2f*)(L + 16);
            v2f b2 = *(const v2f*)(L + 32);
            v2f b3 = *(const v2f*)(L + 48);
            acc0 = __builtin_amdgcn_wmma_f32_16x16x4_f32(false, a, false, b0, (short)0, acc0, false, false);
            acc1 = __builtin_amdgcn_wmma_f32_16x16x4_f32(false, a, false, b1, (short)0, acc1, false, false);
            acc2 = __builtin_amdgcn_wmma_f32_16x16x4_f32(false, a, false, b2, (short)0, acc2, false, false);
            acc3 = __builtin_amdgcn_wmma_f32_16x16x4_f32(false, a, false, b3, (short)0, acc3, false, false);
        }
        __syncthreads();
    }

    if (mt < MT) {
        // C/D layout: VGPR r -> M = r + 8*half, N = lane%16
        float* C = T + (size_t)(mtile * 16 + 8 * half) * NO + n0 + mr;
#pragma unroll
        for (int r = 0; r < 8; ++r) {
            C[(size_t)r * NO +  0] = acc0[r];
            C[(size_t)r * NO + 16] = acc1[r];
            C[(size_t)r * NO + 32] = acc2[r];
            C[(size_t)r * NO + 48] = acc3[r];
        }
    }
}

// ---------- aggregation ----------
__global__ void k_selfloop(const float* __restrict__ T, const float* __restrict__ dinv,
                           float* __restrict__ agg, int total, int logF) {
    int i = blockIdx.x * blockDim.x + threadIdx.x;
    if (i >= total) return;
    float d = dinv[i >> logF];
    agg[i] = d * d * T[i];
}

// F in {512,128}: one wave handles 128 floats of one edge (float4 per lane)
template <int F>
__global__ void k_edge_agg(const float* __restrict__ T, const int* __restrict__ row,
                           const int* __restrict__ col, const float* __restrict__ nrm,
                           float* agg, int e_cnt) {
    constexpr int WPE = F / 128;         // waves per edge
    const int lane = threadIdx.x & 31;
    const int gw   = (int)((blockIdx.x * blockDim.x + threadIdx.x) >> 5);
    const int e    = gw / WPE;
    const int q    = gw % WPE;
    if (e >= e_cnt) return;              // wave-uniform
    const int r = row[e], c = col[e];
    const float w = nrm[e];
    const int f = q * 128 + lane * 4;
    const float4 v = *(const float4*)(T + (size_t)r * F + f);
    float* dst = agg + (size_t)c * F + f;
    atomAddF(dst + 0, w * v.x);
    atomAddF(dst + 1, w * v.y);
    atomAddF(dst + 2, w * v.z);
    atomAddF(dst + 3, w * v.w);
}

__global__ void k_edge_agg2(const float* __restrict__ T, const int* __restrict__ row,
                            const int* __restrict__ col, const float* __restrict__ nrm,
                            float* out, int e_cnt) {
    int e = blockIdx.x * blockDim.x + threadIdx.x;
    if (e >= e_cnt) return;
    int r = row[e], c = col[e];
    float w = nrm[e];
    float2 v = *(const float2*)(T + 2 * (size_t)r);
    atomAddF(out + 2 * (size_t)c + 0, w * v.x);
    atomAddF(out + 2 * (size_t)c + 1, w * v.y);
}

__global__ void k_bias_act(float* __restrict__ h, const float* __restrict__ b,
                           int total, int fmask, int relu) {
    int i = blockIdx.x * blockDim.x + threadIdx.x;
    if (i >= total) return;
    float v = h[i] + b[i & fmask];
    h[i] = relu ? fmaxf(v, 0.0f) : v;
}

// ---------- layer 2: [N,128] @ [128,2], wave per node ----------
__global__ void k_fc2(const float* __restrict__ h, const float* __restrict__ W2,
                      float* __restrict__ t2, int n) {
    const int lane = threadIdx.x & 31;
    const int node = (int)((blockIdx.x * blockDim.x + threadIdx.x) >> 5);
    if (node >= n) return;
    float4 v = *(const float4*)(h + (size_t)node * 128 + lane * 4);
    const float* w = W2 + lane * 8;      // rows 4*lane .. 4*lane+3, 2 cols each
    float s0 = v.x * w[0] + v.y * w[2] + v.z * w[4] + v.w * w[6];
    float s1 = v.x * w[1] + v.y * w[3] + v.z * w[5] + v.w * w[7];
#pragma unroll
    for (int off = 16; off > 0; off >>= 1) {
        s0 += __shfl_down(s0, off, 32);
        s1 += __shfl_down(s1, off, 32);
    }
    if (lane == 0) {
        t2[2 * (size_t)node + 0] = s0;
        t2[2 * (size_t)node + 1] = s1;
    }
}

extern "C" void kernel_launch(void* const* d_in, const int* in_sizes, int n_in,
                              void* d_out, int out_size, void* d_ws, size_t ws_size,
                              hipStream_t stream) {
    const float*     x  = (const float*)d_in[0];
    const long long* ei = (const long long*)d_in[1];
    const float*     W0 = (const float*)d_in[2];
    const float*     b0 = (const float*)d_in[3];
    const float*     W1 = (const float*)d_in[4];
    const float*     b1 = (const float*)d_in[5];
    const float*     W2 = (const float*)d_in[6];
    const float*     b2 = (const float*)d_in[7];
    float* out = (float*)d_out;

    const int N = NNODES, E = NEDGES;

    // workspace carve-up (all 16B aligned)
    int*   row32 = (int*)d_ws;
    int*   col32 = row32 + E;
    float* nrm   = (float*)(col32 + E);
    float* dinv  = nrm + E;                  // holds deg, then rsqrt in place
    float* t0    = dinv + N;                 // N x 512
    float* h1    = t0 + (size_t)N * 512;     // N x 512 (agg0, then relu'd)
    float* t1    = h1 + (size_t)N * 512;     // N x 128
    float* h2    = t1 + (size_t)N * 128;     // N x 128
    float* t2    = h2 + (size_t)N * 128;     // N x 2

    // --- normalization ---
    k_init_deg <<<(N + 255) / 256, 256, 0, stream>>>(dinv, N);
    k_count_deg<<<(E + 255) / 256, 256, 0, stream>>>(ei, row32, col32, dinv, E);
    k_dinv     <<<(N + 255) / 256, 256, 0, stream>>>(dinv, N);
    k_norm     <<<(E + 255) / 256, 256, 0, stream>>>(row32, col32, dinv, nrm, E);

    const int MT = N / 16;                   // 3125 (N is a multiple of 16)

    // --- layer 0: 1024 -> 512 ---
    dim3 g0((MT + 7) / 8, 512 / 64);
    k_gemm_wmma<<<g0, 256, 0, stream>>>(x, W0, t0, N, 1024, 512);
    k_selfloop <<<(N * 512 + 255) / 256, 256, 0, stream>>>(t0, dinv, h1, N * 512, 9);
    k_edge_agg<512><<<(E * 4 + 7) / 8, 256, 0, stream>>>(t0, row32, col32, nrm, h1, E);
    k_bias_act <<<(N * 512 + 255) / 256, 256, 0, stream>>>(h1, b0, N * 512, 511, 1);

    // --- layer 1: 512 -> 128 ---
    dim3 g1((MT + 7) / 8, 128 / 64);
    k_gemm_wmma<<<g1, 256, 0, stream>>>(h1, W1, t1, N, 512, 128);
    k_selfloop <<<(N * 128 + 255) / 256, 256, 0, stream>>>(t1, dinv, h2, N * 128, 7);
    k_edge_agg<128><<<(E + 7) / 8, 256, 0, stream>>>(t1, row32, col32, nrm, h2, E);
    k_bias_act <<<(N * 128 + 255) / 256, 256, 0, stream>>>(h2, b1, N * 128, 127, 1);

    // --- layer 2: 128 -> 2 ---
    k_fc2      <<<(N + 7) / 8, 256, 0, stream>>>(h2, W2, t2, N);
    k_selfloop <<<(N * 2 + 255) / 256, 256, 0, stream>>>(t2, dinv, out, N * 2, 1);
    k_edge_agg2<<<(E + 255) / 256, 256, 0, stream>>>(t2, row32, col32, nrm, out, E);
    k_bias_act <<<(N * 2 + 255) / 256, 256, 0, stream>>>(out, b2, N * 2, 1, 0);
}